// RobustSum_35811437314535
// MI455X (gfx1250) — compile-verified
//
#include <hip/hip_runtime.h>
#include <hip/hip_bf16.h>

typedef __attribute__((ext_vector_type(2))) float v2f;
typedef __attribute__((ext_vector_type(8))) float v8f;

#define DIN   1024
#define DOUT  1024
#define BATCH 128
#define EPS   0.001f
#define NORM_EPS 1e-12f
#define BTILE 4

// ---------------------------------------------------------------------------
// Kernel 1: z0 = x @ W, full fp32 via V_WMMA_F32_16X16X4_F32.
// One wave (32 threads) per 16x16 tile of z. 8 x 64 = 512 tiles.
// A layout (16x4 f32): lanes 0-15 -> M=lane, K={k0,k0+1}; lanes 16-31 -> K={k0+2,k0+3}
// B layout (4x16 f32): lanes 0-15 -> N=lane, K={k0,k0+1}; lanes 16-31 -> K={k0+2,k0+3}
// C/D layout: vgpr r -> row (r + 8*(lane>=16)), col (lane&15)
// ---------------------------------------------------------------------------
__global__ __launch_bounds__(32) void robustsum_seed_gemm(
    const float* __restrict__ X,   // (BATCH, DIN)
    const float* __restrict__ W,   // (DIN, DOUT)
    float* __restrict__ Z)         // (BATCH, DOUT)
{
    const int tile = blockIdx.x;      // 0..511
    const int m0 = (tile >> 6) * 16;  // 8 M-tiles
    const int n0 = (tile & 63) * 16;  // 64 N-tiles
    const int lane = threadIdx.x;
    const int half = lane >> 4;       // 0: K pair {0,1}, 1: K pair {2,3}
    const int l15  = lane & 15;

    const float* xrow = X + (size_t)(m0 + l15) * DIN;  // A row m = l15
    const float* wcol = W + n0 + l15;                  // B col n = l15

    v8f c = {};
#pragma unroll 4
    for (int k0 = 0; k0 < DIN; k0 += 4) {
        const int ka = k0 + half * 2;
        v2f a, b;
        a.x = xrow[ka];
        a.y = xrow[ka + 1];
        b.x = wcol[(size_t)ka * DOUT];
        b.y = wcol[(size_t)(ka + 1) * DOUT];
        // 8 args: (neg_a, A, neg_b, B, c_mod, C, reuse_a, reuse_b)
        c = __builtin_amdgcn_wmma_f32_16x16x4_f32(
            false, a, false, b, (short)0, c, false, false);
    }

    float* zout = Z + (size_t)(m0 + half * 8) * DOUT + n0 + l15;
#pragma unroll
    for (int r = 0; r < 8; ++r)
        zout[(size_t)r * DOUT] = c[r];
}

// ---------------------------------------------------------------------------
// Kernel 2: 3 IRLS sweeps. Block = 128 threads = 128 output cols, BTILE=4
// batch rows per block. x rows cached in LDS (16 KB); each W[i][o] load is
// reused across the 4 batch rows. z / Σw / Σ(w*xw) stay in registers.
// Since w > 0, Σ|w| == Σw  ->  z = D * Σ(w*xw) / max(Σw, 1e-12).
// ---------------------------------------------------------------------------
__global__ __launch_bounds__(128) void robustsum_irls(
    const float* __restrict__ X,    // (BATCH, DIN)
    const float* __restrict__ W,    // (DIN, DOUT)
    const float* __restrict__ Z0,   // (BATCH, DOUT) seed
    float* __restrict__ Out)        // (BATCH, DOUT)
{
    __shared__ float xs[BTILE][DIN];

    const int o  = blockIdx.x * 128 + threadIdx.x;  // output column
    const int b0 = blockIdx.y * BTILE;              // first batch row

    // cooperative load of the BTILE x-rows into LDS
    for (int idx = threadIdx.x; idx < BTILE * DIN; idx += 128) {
        const int bb = idx >> 10;
        const int ii = idx & (DIN - 1);
        xs[bb][ii] = X[(size_t)(b0 + bb) * DIN + ii];
    }
    __syncthreads();

    float z[BTILE];
#pragma unroll
    for (int bb = 0; bb < BTILE; ++bb)
        z[bb] = Z0[(size_t)(b0 + bb) * DOUT + o];

    const float invD = 1.0f / (float)DIN;
    const float* wp = W + o;

    for (int it = 0; it < 3; ++it) {
        float zc[BTILE], sw[BTILE], swx[BTILE];
#pragma unroll
        for (int bb = 0; bb < BTILE; ++bb) {
            zc[bb] = z[bb] * invD;
            sw[bb] = 0.0f;
            swx[bb] = 0.0f;
        }

#pragma unroll 4
        for (int i = 0; i < DIN; ++i) {
            const float wv = wp[(size_t)i * DOUT];   // one load, 4x reuse
#pragma unroll
            for (int bb = 0; bb < BTILE; ++bb) {
                const float xw = xs[bb][i] * wv;
                const float d  = __builtin_fabsf(xw - zc[bb]) + EPS;
                const float wg = __builtin_amdgcn_rcpf(d);   // v_rcp_f32
                sw[bb]  += wg;
                swx[bb] += wg * xw;
            }
        }

#pragma unroll
        for (int bb = 0; bb < BTILE; ++bb) {
            const float denom = fmaxf(sw[bb], NORM_EPS);
            z[bb] = (float)DIN * swx[bb] / denom;    // epilogue: exact div
        }
    }

#pragma unroll
    for (int bb = 0; bb < BTILE; ++bb)
        Out[(size_t)(b0 + bb) * DOUT + o] = z[bb];
}

// ---------------------------------------------------------------------------
extern "C" void kernel_launch(void* const* d_in, const int* in_sizes, int n_in,
                              void* d_out, int out_size, void* d_ws, size_t ws_size,
                              hipStream_t stream) {
    const float* x = (const float*)d_in[0];   // (128, 1024)
    const float* w = (const float*)d_in[1];   // (1024, 1024)
    float* out = (float*)d_out;               // (128, 1024)
    float* z0  = (float*)d_ws;                // seed z, 512 KB scratch

    // seed GEMM: 512 16x16 tiles, one wave each
    robustsum_seed_gemm<<<dim3(512), dim3(32), 0, stream>>>(x, w, z0);

    // IRLS: 8 column-tiles x 32 row-groups
    robustsum_irls<<<dim3(DOUT / 128, BATCH / BTILE), dim3(128), 0, stream>>>(
        x, w, z0, out);
}